// CRFLayer_3762391351716
// MI455X (gfx1250) — compile-verified
//
#include <hip/hip_runtime.h>
#include <hip/hip_bf16.h>
#include <stdint.h>

// Problem constants (match reference setup_inputs()).
#define S_LEN   32768
#define D_DIM   1024
#define L_LAB   64
#define CHUNK   128
#define NCHUNK  256          // ceil((S_LEN-1)/CHUNK)
#define START_TAG 0
#define END_TAG   63
#define MAX_NEG (-1.0e30f)

typedef __attribute__((ext_vector_type(2))) float v2f;
typedef __attribute__((ext_vector_type(8))) float v8f;

// ---------------------------------------------------------------------------
// K1: emissions[S,64] = feats[S,1024] @ W.T + b   via V_WMMA_F32_16X16X4_F32.
// One wave per 16x16 output tile; 4 waves per block cover N=64.
// A 16x4 f32 layout: lanes 0-15 -> M=lane,K={k0,k0+1}; lanes 16-31 -> K={k0+2,k0+3}.
// B 4x16 f32 layout mirrors it with N across lanes.
// ---------------------------------------------------------------------------
__global__ __launch_bounds__(128) void k_emis(const float* __restrict__ feats,
                                              const float* __restrict__ W,
                                              const float* __restrict__ b,
                                              float* __restrict__ E) {
  const int wave = threadIdx.x >> 5;
  const int lane = threadIdx.x & 31;
  const int m0 = blockIdx.x * 16;
  const int n0 = wave * 16;
  const int half = lane >> 4;     // 0 or 1
  const int sub  = lane & 15;
  const int kh = half * 2;
  const float* arow = feats + (size_t)(m0 + sub) * D_DIM + kh; // A: M=sub
  const float* brow = W     + (size_t)(n0 + sub) * D_DIM + kh; // B: N=sub, B[k][n]=W[n][k]
  v8f acc = {};
  for (int k0 = 0; k0 < D_DIM; k0 += 4) {
    v2f a  = *reinterpret_cast<const v2f*>(arow + k0);
    v2f bb = *reinterpret_cast<const v2f*>(brow + k0);
    acc = __builtin_amdgcn_wmma_f32_16x16x4_f32(false, a, false, bb,
                                                (short)0, acc, false, false);
  }
  const float bias = b[n0 + sub];
  // C/D layout: VGPR r holds M=r (lanes 0-15) and M=r+8 (lanes 16-31), N=sub.
  #pragma unroll
  for (int r = 0; r < 8; ++r) {
    int row = r + half * 8;
    E[(size_t)(m0 + row) * L_LAB + (n0 + sub)] = acc[r] + bias;
  }
}

// ---------------------------------------------------------------------------
// K2 (phase 1): per-chunk condensed max-plus matrix.
//   P_c such that scores[e_c] = P_c (x) scores[s_c]   (max-plus)
//   step update: P'[j,k] = e_t[j] + max_i( T[j,i] + P[i,k] )
// 1024 threads: thread owns k = tid&63 and j in {4*jb..4*jb+3}, jb = tid>>6.
// P reads lane-consecutive (bank-conflict-free); T reads wave-uniform b128.
// ---------------------------------------------------------------------------
__global__ __launch_bounds__(1024) void k_phase1(const float* __restrict__ E,
                                                 const float* __restrict__ trans,
                                                 float* __restrict__ Pout) {
  __shared__ float Pbuf[2][64 * 64];
  __shared__ float Tt[64 * 64];     // Tt[i*64+j] = trans[j*64+i]
  const int tid  = threadIdx.x;
  const int c    = blockIdx.x;
  const int kcol = tid & 63;
  const int jb   = (tid >> 6) * 4;  // 0,4,...,60 ; uniform within a wave32

  for (int o = tid; o < 4096; o += 1024)
    Tt[o] = trans[(o & 63) * 64 + (o >> 6)];

  const int sc = c * CHUNK;
  const int ec = min(sc + CHUNK, S_LEN - 1);

  // Init with first step t0 = sc+1: P[j,k] = e_{t0}[j] + T[j,k]
  #pragma unroll
  for (int r = 0; r < 4; ++r) {
    int j = jb + r;
    Pbuf[0][j * 64 + kcol] = E[(size_t)(sc + 1) * L_LAB + j] + trans[j * 64 + kcol];
  }
  __syncthreads();

  int p = 0;
  for (int t = sc + 2; t <= ec; ++t) {
    const float e0 = E[(size_t)t * L_LAB + jb + 0];
    const float e1 = E[(size_t)t * L_LAB + jb + 1];
    const float e2 = E[(size_t)t * L_LAB + jb + 2];
    const float e3 = E[(size_t)t * L_LAB + jb + 3];
    float m0 = MAX_NEG, m1 = MAX_NEG, m2 = MAX_NEG, m3 = MAX_NEG;
    const float* Pa = Pbuf[p];
    #pragma unroll 8
    for (int i = 0; i < 64; ++i) {
      float pv = Pa[i * 64 + kcol];
      float4 tv = *reinterpret_cast<const float4*>(&Tt[i * 64 + jb]);
      m0 = fmaxf(m0, tv.x + pv);
      m1 = fmaxf(m1, tv.y + pv);
      m2 = fmaxf(m2, tv.z + pv);
      m3 = fmaxf(m3, tv.w + pv);
    }
    float* Pn = Pbuf[p ^ 1];
    Pn[(jb + 0) * 64 + kcol] = e0 + m0;
    Pn[(jb + 1) * 64 + kcol] = e1 + m1;
    Pn[(jb + 2) * 64 + kcol] = e2 + m2;
    Pn[(jb + 3) * 64 + kcol] = e3 + m3;
    p ^= 1;
    __syncthreads();
  }

  // Each thread re-reads only values it wrote itself (no sync needed).
  #pragma unroll
  for (int r = 0; r < 4; ++r) {
    int j = jb + r;
    Pout[(size_t)c * 4096 + j * 64 + kcol] = Pbuf[p][j * 64 + kcol];
  }
}

// ---------------------------------------------------------------------------
// K3: sequential combine over chunks -> boundary score vectors SB[c][64].
// SB[c] = scores at t = c*CHUNK (entry of chunk c); SB[NCHUNK] = final.
// ---------------------------------------------------------------------------
__global__ __launch_bounds__(64) void k_combine(const float* __restrict__ E,
                                                const float* __restrict__ trans,
                                                const float* __restrict__ P,
                                                float* __restrict__ SB) {
  __shared__ float s_lds[64];
  const int j = threadIdx.x;
  float s = trans[j * 64 + START_TAG] + E[j];   // init_scores
  SB[j] = s;
  s_lds[j] = s;
  __syncthreads();
  for (int c = 0; c < NCHUNK; ++c) {
    const float* Pr = P + (size_t)c * 4096 + j * 64;
    float m = MAX_NEG;
    #pragma unroll 8
    for (int k = 0; k < 64; ++k)
      m = fmaxf(m, Pr[k] + s_lds[k]);
    __syncthreads();
    s_lds[j] = m;
    SB[(c + 1) * 64 + j] = m;
    __syncthreads();
  }
}

// ---------------------------------------------------------------------------
// K4: exact parallel replay per chunk -> back-pointers (u8) + final scores.
// Thread j keeps its transition row T[j][*] in 64 VGPRs; scores in LDS.
// Strict '>' keeps the FIRST max index, matching jnp.argmax tie-breaking.
// ---------------------------------------------------------------------------
__global__ __launch_bounds__(64) void k_replay(const float* __restrict__ E,
                                               const float* __restrict__ trans,
                                               const float* __restrict__ SB,
                                               uint8_t* __restrict__ BP,
                                               float* __restrict__ FS) {
  __shared__ float sbuf[2][64];
  const int j = threadIdx.x;
  const int c = blockIdx.x;
  float Trow[64];
  #pragma unroll
  for (int i = 0; i < 64; ++i) Trow[i] = trans[j * 64 + i];
  const int sc = c * CHUNK;
  const int ec = min(sc + CHUNK, S_LEN - 1);
  sbuf[0][j] = SB[c * 64 + j];
  __syncthreads();
  int p = 0;
  float cur = 0.0f;
  for (int t = sc + 1; t <= ec; ++t) {
    const float e = E[(size_t)t * L_LAB + j];
    float m = MAX_NEG;
    int am = 0;
    const float* sv = sbuf[p];
    #pragma unroll
    for (int i = 0; i < 64; ++i) {
      float v = sv[i] + Trow[i];
      if (v > m) { m = v; am = i; }
    }
    cur = e + m;
    sbuf[p ^ 1][j] = cur;
    BP[(size_t)(t - 1) * 64 + j] = (uint8_t)am;
    p ^= 1;
    __syncthreads();
  }
  if (c == NCHUNK - 1) FS[j] = cur;   // scores at t = S_LEN-1
}

// ---------------------------------------------------------------------------
// K5a: per-chunk back-pointer map composition (parallel backtrack, phase 1).
// F_c[x] = tag at t=s_c given tag x at t=e_c.
// ---------------------------------------------------------------------------
__global__ __launch_bounds__(64) void k_compose(const uint8_t* __restrict__ BP,
                                                uint8_t* __restrict__ F) {
  __shared__ uint8_t bp_lds[CHUNK * 64];
  const int tid = threadIdx.x;
  const int c = blockIdx.x;
  const int sc = c * CHUNK;
  const int ec = min(sc + CHUNK, S_LEN - 1);
  const int nsteps = ec - sc;
  const uint32_t* src = reinterpret_cast<const uint32_t*>(BP + (size_t)sc * 64);
  uint32_t* dst = reinterpret_cast<uint32_t*>(bp_lds);
  for (int o = tid; o < nsteps * 16; o += 64) dst[o] = src[o];
  __syncthreads();
  int m = tid;
  for (int t = ec; t > sc; --t)
    m = bp_lds[(t - 1 - sc) * 64 + m];
  F[c * 64 + tid] = (uint8_t)m;
}

// ---------------------------------------------------------------------------
// K5b: final score/argmax + 256-step boundary-tag walk (BT[c] = tag at e_c).
// ---------------------------------------------------------------------------
__global__ __launch_bounds__(64) void k_reduce_walk(const float* __restrict__ FS,
                                                    const float* __restrict__ trans,
                                                    const uint8_t* __restrict__ F,
                                                    uint8_t* __restrict__ BT,
                                                    float* __restrict__ out_score) {
  __shared__ float fsl[64];
  __shared__ uint8_t Fl[NCHUNK * 64];   // 16 KB
  const int j = threadIdx.x;
  fsl[j] = FS[j] + trans[END_TAG * 64 + j];
  const uint32_t* src = reinterpret_cast<const uint32_t*>(F);
  uint32_t* dst = reinterpret_cast<uint32_t*>(Fl);
  for (int o = j; o < NCHUNK * 16; o += 64) dst[o] = src[o];
  __syncthreads();
  if (j == 0) {
    float best = fsl[0];
    int bi = 0;
    for (int i = 1; i < 64; ++i)
      if (fsl[i] > best) { best = fsl[i]; bi = i; }
    out_score[0] = best;
    int y = bi;
    BT[NCHUNK - 1] = (uint8_t)y;
    for (int c = NCHUNK - 1; c >= 1; --c) {
      y = Fl[c * 64 + y];
      BT[c - 1] = (uint8_t)y;
    }
  }
}

// ---------------------------------------------------------------------------
// K5c: per-chunk tag expansion; tags staged in LDS, stored coalesced as f32.
// ---------------------------------------------------------------------------
__global__ __launch_bounds__(64) void k_expand(const uint8_t* __restrict__ BP,
                                               const uint8_t* __restrict__ BT,
                                               float* __restrict__ out_tags) {
  __shared__ uint8_t bp_lds[CHUNK * 64];
  __shared__ int16_t tag_lds[CHUNK + 1];
  const int tid = threadIdx.x;
  const int c = blockIdx.x;
  const int sc = c * CHUNK;
  const int ec = min(sc + CHUNK, S_LEN - 1);
  const int nsteps = ec - sc;
  const uint32_t* src = reinterpret_cast<const uint32_t*>(BP + (size_t)sc * 64);
  uint32_t* dst = reinterpret_cast<uint32_t*>(bp_lds);
  for (int o = tid; o < nsteps * 16; o += 64) dst[o] = src[o];
  __syncthreads();
  if (tid == 0) {
    int y = BT[c];
    tag_lds[nsteps] = (int16_t)y;              // tag at t = ec
    for (int t = ec; t > sc; --t) {
      y = bp_lds[(t - 1 - sc) * 64 + y];
      tag_lds[t - 1 - sc] = (int16_t)y;        // tag at t-1
    }
  }
  __syncthreads();
  for (int L = tid; L <= nsteps; L += 64)
    out_tags[sc + L] = (float)tag_lds[L];
}

// ---------------------------------------------------------------------------
// Host-side launcher. Workspace layout (bytes, all 64-aligned):
//   E  @ 0          : S*64*4        = 8,388,608
//   P  @ 8,388,608  : NCHUNK*4096*4 = 4,194,304
//   SB @ 12,582,912 : (NCHUNK+1)*64*4
//   FS @ 12,648,704 : 64*4 (padded to 256)
//   BP @ 12,648,960 : (S-1)*64 u8
//   F  @ 14,746,048 : NCHUNK*64 u8
//   BT @ 14,762,432 : NCHUNK u8
// ---------------------------------------------------------------------------
extern "C" void kernel_launch(void* const* d_in, const int* in_sizes, int n_in,
                              void* d_out, int out_size, void* d_ws, size_t ws_size,
                              hipStream_t stream) {
  (void)in_sizes; (void)n_in; (void)out_size; (void)ws_size;
  const float* feats = (const float*)d_in[0];
  const float* W     = (const float*)d_in[1];
  const float* bias  = (const float*)d_in[2];
  const float* trans = (const float*)d_in[3];
  float* out = (float*)d_out;

  char* ws = (char*)d_ws;
  float*   E  = (float*)(ws + 0);
  float*   P  = (float*)(ws + 8388608u);
  float*   SB = (float*)(ws + 12582912u);
  float*   FS = (float*)(ws + 12648704u);
  uint8_t* BP = (uint8_t*)(ws + 12648960u);
  uint8_t* F  = (uint8_t*)(ws + 14746048u);
  uint8_t* BT = (uint8_t*)(ws + 14762432u);

  k_emis<<<S_LEN / 16, 128, 0, stream>>>(feats, W, bias, E);
  k_phase1<<<NCHUNK, 1024, 0, stream>>>(E, trans, P);
  k_combine<<<1, 64, 0, stream>>>(E, trans, P, SB);
  k_replay<<<NCHUNK, 64, 0, stream>>>(E, trans, SB, BP, FS);
  k_compose<<<NCHUNK, 64, 0, stream>>>(BP, F);
  k_reduce_walk<<<1, 64, 0, stream>>>(FS, trans, F, BT, out);
  k_expand<<<NCHUNK, 64, 0, stream>>>(BP, BT, out + 1);
}